// Mamba2_50646254355296
// MI455X (gfx1250) — compile-verified
//
#include <hip/hip_runtime.h>
#include <math.h>

// ---------------------------------------------------------------------------
// Mamba2 forward for MI455X (gfx1250, wave32).
// GEMMs: TDM (tensor_load_to_lds) -> LDS double buffer -> V_WMMA_F32_16X16X4_F32
// with software-pipelined LDS fragment loads inside the K-tile.
// ---------------------------------------------------------------------------

typedef __attribute__((ext_vector_type(2))) float v2f;
typedef __attribute__((ext_vector_type(8))) float v8f;
typedef __attribute__((ext_vector_type(4))) unsigned int u32x4;
typedef __attribute__((ext_vector_type(4))) int i32x4;
typedef __attribute__((ext_vector_type(8))) int i32x8;

#define D_MODEL   1024
#define D_STATE   64
#define D_HEAD    64
#define N_HEADS   32
#define D_CONV    4
#define D_INNER   2048          // N_HEADS * D_HEAD
#define XBC_DIM   2176          // D_INNER + 2*D_STATE
#define PROJ_DIM  4256          // 2*D_INNER + 2*D_STATE + N_HEADS
#define B_SZ      2
#define L_SEQ     1024
#define ROWS      (B_SZ * L_SEQ)

// GEMM tiling
#define BM 128
#define BN 128
#define BK 16
#define AS_STRIDE (BK + 2)      // TDM pads 2 dwords per 16-dword row -> no LDS bank conflicts

// LDS byte address of a __shared__ object (generic -> addrspace(3) -> offset)
__device__ __forceinline__ unsigned lds_addr_of(const void* p)
{
    return (unsigned)(unsigned long long)(__attribute__((address_space(3))) const void*)p;
}

// Issue one 2D TDM load: global (row-major, 4-byte elems) tile -> LDS.
// flagsW0 carries data_size + optional LDS padding config (D# group1 word0).
__device__ __forceinline__ void tdm_load_2d(
    unsigned ldsAddr, unsigned long long gaddr, unsigned flagsW0,
    unsigned tensor_d0, unsigned tensor_d1,
    unsigned tile_d0, unsigned tile_d1, unsigned long long stride_d0)
{
    u32x4 g0;
    g0.x = 1u;                                        // count=1, user descriptor
    g0.y = ldsAddr;                                   // lds_addr (bytes)
    g0.z = (unsigned)gaddr;                           // global_addr[31:0]
    g0.w = (unsigned)(gaddr >> 32) | (2u << 30);      // global_addr[56:32] | type=2

    i32x8 g1;
    g1[0] = (int)flagsW0;                                         // wg_mask=0, data_size, pad cfg
    g1[1] = (int)((tensor_d0 & 0xFFFFu) << 16);                   // atomic_addr=0 | tensor_dim0 lo16
    g1[2] = (int)((tensor_d0 >> 16) | ((tensor_d1 & 0xFFFFu) << 16));
    g1[3] = (int)((tensor_d1 >> 16) | (tile_d0 << 16));           // tensor_dim1 hi | tile_dim0
    g1[4] = (int)(tile_d1 & 0xFFFFu);                             // tile_dim1 | tile_dim2=0
    g1[5] = (int)(unsigned)stride_d0;                             // tensor_dim0_stride lo32
    g1[6] = (int)(unsigned)((stride_d0 >> 32) & 0xFFFFu);         // stride hi16 | dim1_stride lo=0
    g1[7] = 0;

    i32x4 z4 = {0, 0, 0, 0};
#if defined(__clang_major__) && (__clang_major__ >= 23)
    i32x8 z8 = {0, 0, 0, 0, 0, 0, 0, 0};
    __builtin_amdgcn_tensor_load_to_lds(g0, g1, z4, z4, z8, 0);
#else
    __builtin_amdgcn_tensor_load_to_lds(g0, g1, z4, z4, 0);
#endif
}

// ---------------------------------------------------------------------------
// C[M,N] = A[M,K] @ B[K,N], all row-major fp32, WMMA f32 16x16x4.
// Block: 256 threads (8 waves), tile 128x128, BK=16, double-buffered LDS via TDM.
// Waves: 4 in M x 2 in N; wave tile 32x64 = 2x4 accumulators of 16x16.
// Inside a K-tile, LDS fragments are double-buffered in registers so ds_load
// latency for step kk+4 hides under the 8 WMMAs of step kk.
// Requires M % 128 == 0, K % 16 == 0; N boundary handled (TDM OOB reads -> 0,
// stores guarded).
// ---------------------------------------------------------------------------
__global__ __launch_bounds__(256) void gemm_f32_wmma_tdm(
    const float* __restrict__ A, const float* __restrict__ B,
    float* __restrict__ C, int M, int N, int K, int lda, int ldb, int ldc)
{
    __shared__ float As[2][BM * AS_STRIDE];
    __shared__ float Bs[2][BK * BN];

    const int tid  = threadIdx.x;
    const int lane = tid & 31;
    const int wave = tid >> 5;
    const int wm   = wave & 3;                  // 0..3  (M direction)
    const int wn   = wave >> 2;                 // 0..1  (N direction)
    const int rowBase = blockIdx.y * BM;
    const int colBase = blockIdx.x * BN;

    // WMMA f32 16x16x4 fragment lane mapping
    const int m0 = lane & 15;
    const int kb = (lane >> 4) << 1;            // 0 or 2
    const int n0 = lane & 15;

    v8f acc[2][4] = {};

    // D# group1 word0: data_size=2 (4B); A adds LDS padding 2 dwords / 16 dwords
    const unsigned flagsA = (2u << 16) | (1u << 20) | (3u << 22) | (1u << 25);
    const unsigned flagsB = (2u << 16);

    const unsigned ldsA0 = lds_addr_of(&As[0][0]);
    const unsigned ldsA1 = lds_addr_of(&As[1][0]);
    const unsigned ldsB0 = lds_addr_of(&Bs[0][0]);
    const unsigned ldsB1 = lds_addr_of(&Bs[1][0]);

    unsigned long long gA = (unsigned long long)(A + (long)rowBase * lda);
    unsigned long long gB = (unsigned long long)(B + (long)colBase);
    const unsigned long long stepA = (unsigned long long)BK * 4ull;
    const unsigned long long stepB = (unsigned long long)BK * (unsigned long long)ldb * 4ull;
    const unsigned remN = (unsigned)(N - colBase);   // TDM bounds-checks dim0 against this

    const int NT = K / BK;

    // Prologue: stage tile 0 into buffer 0
    if (wave == 0) {
        tdm_load_2d(ldsA0, gA, flagsA, BK, BM, BK, BM, (unsigned long long)lda);
        tdm_load_2d(ldsB0, gB, flagsB, remN, BK, BN, BK, (unsigned long long)ldb);
    }

    for (int t = 0; t < NT; ++t) {
        const int cur = t & 1;
        if (wave == 0)
            __builtin_amdgcn_s_wait_tensorcnt(0);   // tile t landed in LDS
        __syncthreads();                            // release block; prior compute done

        if (wave == 0 && (t + 1) < NT) {            // prefetch tile t+1 into other buffer
            gA += stepA;
            gB += stepB;
            tdm_load_2d(cur ? ldsA0 : ldsA1, gA, flagsA, BK, BM, BK, BM,
                        (unsigned long long)lda);
            tdm_load_2d(cur ? ldsB0 : ldsB1, gB, flagsB, remN, BK, BN, BK,
                        (unsigned long long)ldb);
        }

        const float* __restrict__ Asb = &As[cur][0];
        const float* __restrict__ Bsb = &Bs[cur][0];
        const int ra0 = (wm * 32 + m0) * AS_STRIDE + kb;
        const int ra1 = ra0 + 16 * AS_STRIDE;
        const int cb  = wn * 64 + n0;

        // Register double-buffered fragments: slot s computes kk, slot s^1
        // is being filled for kk+4 while the WMMAs run.
        v2f a0[2], a1[2], bf[2][4];
        a0[0] = *(const v2f*)(Asb + ra0);
        a1[0] = *(const v2f*)(Asb + ra1);
        {
            const float* Bk = Bsb + kb * BN;
#pragma unroll
            for (int j = 0; j < 4; ++j) {
                bf[0][j].x = Bk[cb + j * 16];
                bf[0][j].y = Bk[BN + cb + j * 16];
            }
        }

#pragma unroll
        for (int kk = 0; kk < BK; kk += 4) {
            const int s  = (kk >> 2) & 1;
            const int ns = s ^ 1;
            if (kk + 4 < BK) {                      // prefetch next k-step's fragments
                a0[ns] = *(const v2f*)(Asb + ra0 + kk + 4);
                a1[ns] = *(const v2f*)(Asb + ra1 + kk + 4);
                const float* Bk = Bsb + (kk + 4 + kb) * BN;
#pragma unroll
                for (int j = 0; j < 4; ++j) {
                    bf[ns][j].x = Bk[cb + j * 16];
                    bf[ns][j].y = Bk[BN + cb + j * 16];
                }
            }
#pragma unroll
            for (int j = 0; j < 4; ++j) {
                acc[0][j] = __builtin_amdgcn_wmma_f32_16x16x4_f32(
                    false, a0[s], false, bf[s][j], (short)0, acc[0][j], false, false);
                acc[1][j] = __builtin_amdgcn_wmma_f32_16x16x4_f32(
                    false, a1[s], false, bf[s][j], (short)0, acc[1][j], false, false);
            }
        }
        __syncthreads();                            // done reading buffer `cur`
    }

    // Store: VGPR r holds M = r + 8*(lane>=16); N = lane&15 within 16-col sub-tile
    const int mhalf = (lane >> 4) << 3;
#pragma unroll
    for (int i = 0; i < 2; ++i) {
        const int mb = rowBase + wm * 32 + i * 16 + mhalf;
#pragma unroll
        for (int j = 0; j < 4; ++j) {
            const int nn = colBase + wn * 64 + j * 16 + n0;
            if (nn < N) {
#pragma unroll
                for (int r = 0; r < 8; ++r)
                    C[(long)(mb + r) * ldc + nn] = acc[i][j][r];
            }
        }
    }
}

// ---------------------------------------------------------------------------
// Depthwise causal conv (4 taps) + bias + SiLU on the xBC slice of xz.
// ---------------------------------------------------------------------------
__global__ __launch_bounds__(256) void conv_silu_kernel(
    const float* __restrict__ xz, const float* __restrict__ conv_w,
    const float* __restrict__ conv_b, float* __restrict__ xc)
{
    const int idx = blockIdx.x * blockDim.x + threadIdx.x;
    if (idx >= ROWS * XBC_DIM) return;
    const int c   = idx % XBC_DIM;
    const int row = idx / XBC_DIM;
    const int l   = row % L_SEQ;
    const int b   = row / L_SEQ;

    const float* w = conv_w + c * D_CONV;
    float acc = conv_b[c];
#pragma unroll
    for (int k = 0; k < D_CONV; ++k) {
        const int lp = l - (D_CONV - 1) + k;
        if (lp >= 0)
            acc += w[k] * xz[(long)(b * L_SEQ + lp) * PROJ_DIM + D_INNER + c];
    }
    acc = acc / (1.0f + expf(-acc));            // silu
    xc[(long)row * XBC_DIM + c] = acc;
}

// ---------------------------------------------------------------------------
// Sequential SSM scan. One workgroup per (batch, head).
// Thread t: d = t>>2 (0..63), s-chunk = (t&3)*16; 16 state floats in registers.
// ---------------------------------------------------------------------------
__global__ __launch_bounds__(256) void scan_kernel(
    const float* __restrict__ xz, const float* __restrict__ xc,
    const float* __restrict__ A_log, const float* __restrict__ D_param,
    const float* __restrict__ dt_bias, float* __restrict__ ybuf)
{
    const int bh   = blockIdx.x;                // 0..63
    const int b    = bh >> 5;
    const int head = bh & 31;
    const int tid  = threadIdx.x;
    const int d    = tid >> 2;
    const int sq   = (tid & 3) * 16;

    __shared__ float shx[D_HEAD];
    __shared__ float shB[D_STATE];
    __shared__ float shC[D_STATE];

    float h[16];
#pragma unroll
    for (int j = 0; j < 16; ++j) h[j] = 0.0f;

    const float Acoef = -expf(A_log[head]);
    const float dtb   = dt_bias[head];
    const float Dp    = D_param[head];

    for (int t = 0; t < L_SEQ; ++t) {
        const long rowc = (long)(b * L_SEQ + t) * XBC_DIM;
        if (tid < 64)        shx[tid]       = xc[rowc + head * D_HEAD + tid];
        else if (tid < 128)  shB[tid - 64]  = xc[rowc + D_INNER + (tid - 64)];
        else if (tid < 192)  shC[tid - 128] = xc[rowc + D_INNER + D_STATE + (tid - 128)];
        __syncthreads();

        const float dtp = xz[(long)(b * L_SEQ + t) * PROJ_DIM
                             + (D_INNER + XBC_DIM) + head] + dtb;
        const float dt  = (dtp > 20.0f) ? dtp : log1pf(expf(dtp));
        const float dA  = expf(Acoef * dt);
        const float xv  = shx[d];
        const float coef = dt * xv;

        float ysum = 0.0f;
#pragma unroll
        for (int j = 0; j < 16; ++j) {
            h[j] = dA * h[j] + coef * shB[sq + j];
            ysum += h[j] * shC[sq + j];
        }
        ysum += __shfl_xor(ysum, 1);
        ysum += __shfl_xor(ysum, 2);
        if ((tid & 3) == 0)
            ybuf[(long)(b * L_SEQ + t) * D_INNER + head * D_HEAD + d] = ysum + Dp * xv;
        __syncthreads();
    }
}

// ---------------------------------------------------------------------------
// y = y * silu(z); RMSNorm over D_INNER with norm_w. One block per row.
// ---------------------------------------------------------------------------
__global__ __launch_bounds__(256) void gate_norm_kernel(
    const float* __restrict__ xz, const float* __restrict__ norm_w,
    float* __restrict__ ybuf)
{
    const int row = blockIdx.x;
    const int tid = threadIdx.x;

    float g[8];
    float ss = 0.0f;
#pragma unroll
    for (int j = 0; j < 8; ++j) {
        const int idx = tid + j * 256;
        const float z  = xz[(long)row * PROJ_DIM + idx];
        const float yv = ybuf[(long)row * D_INNER + idx];
        const float gv = yv * z / (1.0f + expf(-z));
        g[j] = gv;
        ss += gv * gv;
    }
#pragma unroll
    for (int m = 16; m >= 1; m >>= 1) ss += __shfl_xor(ss, m);

    __shared__ float sred[8];
    if ((tid & 31) == 0) sred[tid >> 5] = ss;
    __syncthreads();
    float tot = 0.0f;
#pragma unroll
    for (int w = 0; w < 8; ++w) tot += sred[w];
    const float scale = rsqrtf(tot / (float)D_INNER + 1e-5f);

#pragma unroll
    for (int j = 0; j < 8; ++j) {
        const int idx = tid + j * 256;
        ybuf[(long)row * D_INNER + idx] = norm_w[idx] * g[j] * scale;
    }
}

// ---------------------------------------------------------------------------
extern "C" void kernel_launch(void* const* d_in, const int* in_sizes, int n_in,
                              void* d_out, int out_size, void* d_ws, size_t ws_size,
                              hipStream_t stream)
{
    const float* x       = (const float*)d_in[0];
    const float* W_in    = (const float*)d_in[1];
    const float* conv_w  = (const float*)d_in[2];
    const float* conv_b  = (const float*)d_in[3];
    const float* A_log   = (const float*)d_in[4];
    const float* D_param = (const float*)d_in[5];
    const float* dt_bias = (const float*)d_in[6];
    const float* norm_w  = (const float*)d_in[7];
    const float* W_out   = (const float*)d_in[8];
    float* out = (float*)d_out;

    // Workspace layout (floats): xz | xc | ybuf  (~70 MB total)
    float* xz   = (float*)d_ws;                          // ROWS x PROJ_DIM
    float* xc   = xz + (size_t)ROWS * PROJ_DIM;          // ROWS x XBC_DIM
    float* ybuf = xc + (size_t)ROWS * XBC_DIM;           // ROWS x D_INNER

    // 1) in-projection GEMM: xz = x @ W_in   (2048x1024 @ 1024x4256)
    {
        dim3 grid((PROJ_DIM + BN - 1) / BN, ROWS / BM);
        gemm_f32_wmma_tdm<<<grid, 256, 0, stream>>>(x, W_in, xz,
                                                    ROWS, PROJ_DIM, D_MODEL,
                                                    D_MODEL, PROJ_DIM, PROJ_DIM);
    }
    // 2) depthwise conv + bias + silu
    {
        const int total = ROWS * XBC_DIM;
        conv_silu_kernel<<<(total + 255) / 256, 256, 0, stream>>>(xz, conv_w,
                                                                  conv_b, xc);
    }
    // 3) sequential SSM scan
    scan_kernel<<<B_SZ * N_HEADS, 256, 0, stream>>>(xz, xc, A_log, D_param,
                                                    dt_bias, ybuf);
    // 4) gate + RMSNorm
    gate_norm_kernel<<<ROWS, 256, 0, stream>>>(xz, norm_w, ybuf);

    // 5) out-projection GEMM: out = ybuf @ W_out  (2048x2048 @ 2048x1024)
    {
        dim3 grid(D_MODEL / BN, ROWS / BM);
        gemm_f32_wmma_tdm<<<grid, 256, 0, stream>>>(ybuf, W_out, out,
                                                    ROWS, D_MODEL, D_INNER,
                                                    D_INNER, D_MODEL, D_MODEL);
    }
}